// m3_KDLoss_14534169329698
// MI455X (gfx1250) — compile-verified
//
#include <hip/hip_runtime.h>

// ---------------------------------------------------------------------------
// KD grouped-CE loss, MI455X (gfx1250, wave32).
//
// Roofline: student_scores is 2048 x 16384 f32 = 128 MiB. The reference's
// G=8 masked logsumexps collapse to ONE streaming pass (per-row max M and
// S = sum exp(x-M)) + an O(B*G) analytic epilogue:
//   lse_i = M + log(S - sum_{j<i} exp(t_j - M)),  ce_i = lse_i - t_i.
// Single pass = 128 MiB / 23.3 TB/s ~ 5.6 us; the naive form is ~45 us.
//
// The exp-sum uses V_WMMA_F32_16X16X4_F32 with B = ones: D[m,n] = sum_k A[m,k],
// i.e. each WMMA reduces 64 f32 values exactly; since B is uniform the A
// layout is irrelevant for the final total. Per-lane sum of the 8 D VGPRs
// covers half the M rows; shfl_xor(16) folds in the other half (C/D layout:
// lanes 0-15 hold M=0..7, lanes 16-31 hold M=8..15).
// ---------------------------------------------------------------------------

typedef float v2f __attribute__((ext_vector_type(2)));
typedef float v8f __attribute__((ext_vector_type(8)));

#define THREADS 256          // 8 waves (wave32) per row
#define VEC_PER_THREAD 16    // 16 x float4 = 64 floats/thread; 256*64 = 16384 = C

__global__ __launch_bounds__(THREADS)
void kd_row_stats_kernel(const float* __restrict__ scores,
                         float* __restrict__ stats,   // [B][2] = {M, S}
                         int C)
{
    const int row = blockIdx.x;
    const int tid = threadIdx.x;
    const float* rowp = scores + (size_t)row * (size_t)C;

    // gfx1250 prefetch path (global_prefetch_b8): warm the far end of the row.
    __builtin_prefetch(rowp + C - THREADS * 4 + tid * 4, 0, 1);

    // ---- load 64 floats/thread, coalesced: thread t, chunk k -> elems
    // [4t + 1024k .. 4t + 1024k + 3] (256 threads * 16B = 4 KiB contiguous).
    float4 r[VEC_PER_THREAD];
    const float4* rp4 = reinterpret_cast<const float4*>(rowp);
#pragma unroll
    for (int k = 0; k < VEC_PER_THREAD; ++k)
        r[k] = rp4[tid + k * THREADS];

    // ---- thread-local max
    float m = r[0].x;
#pragma unroll
    for (int k = 0; k < VEC_PER_THREAD; ++k) {
        m = fmaxf(m, r[k].x);
        m = fmaxf(m, r[k].y);
        m = fmaxf(m, r[k].z);
        m = fmaxf(m, r[k].w);
    }

    // ---- block max reduce
    __shared__ float red[THREADS];
    red[tid] = m;
    __syncthreads();
#pragma unroll
    for (int s = THREADS / 2; s > 0; s >>= 1) {
        if (tid < s) red[tid] = fmaxf(red[tid], red[tid + s]);
        __syncthreads();
    }
    const float M = red[0];
    __syncthreads();

    // ---- exp + WMMA reduction. All 256 threads execute uniformly here
    // (EXEC all-ones, required for WMMA). Two accumulators pipeline the
    // WMMA C-operand dependency chain.
    v8f acc0 = {};
    v8f acc1 = {};
    const v2f ones = {1.0f, 1.0f};
#pragma unroll
    for (int k = 0; k < VEC_PER_THREAD; k += 2) {
        const float4 a = r[k];
        const float4 b = r[k + 1];
        v2f a0 = {__expf(a.x - M), __expf(a.y - M)};
        v2f a1 = {__expf(a.z - M), __expf(a.w - M)};
        v2f b0 = {__expf(b.x - M), __expf(b.y - M)};
        v2f b1 = {__expf(b.z - M), __expf(b.w - M)};
        // D = A * ones + C : exact f32 sum of 64 lane-values per instruction.
        acc0 = __builtin_amdgcn_wmma_f32_16x16x4_f32(false, a0, false, ones,
                                                     (short)0, acc0, false, false);
        acc1 = __builtin_amdgcn_wmma_f32_16x16x4_f32(false, a1, false, ones,
                                                     (short)0, acc1, false, false);
        acc0 = __builtin_amdgcn_wmma_f32_16x16x4_f32(false, b0, false, ones,
                                                     (short)0, acc0, false, false);
        acc1 = __builtin_amdgcn_wmma_f32_16x16x4_f32(false, b1, false, ones,
                                                     (short)0, acc1, false, false);
    }
    v8f acc = acc0 + acc1;

    // per-lane: sum over this half's M rows (D replicated across N, so any
    // lane's column works); fold halves with a cross-16 xor shuffle (wave32).
    float s = acc[0] + acc[1] + acc[2] + acc[3] + acc[4] + acc[5] + acc[6] + acc[7];
    s += __shfl_xor(s, 16, 32);   // now s == this wave's 2048-element exp-sum

    __shared__ float wsum[THREADS / 32];
    const int wave = tid >> 5;
    const int lane = tid & 31;
    if (lane == 0) wsum[wave] = s;
    __syncthreads();
    if (tid == 0) {
        float S = 0.0f;
#pragma unroll
        for (int w = 0; w < THREADS / 32; ++w) S += wsum[w];
        stats[row * 2 + 0] = M;
        stats[row * 2 + 1] = S;
    }
}

// ---------------------------------------------------------------------------
// Epilogue: O(B*G) work, single block, exact masked-lse recurrence.
// ---------------------------------------------------------------------------
__global__ __launch_bounds__(256)
void kd_loss_finalize_kernel(const float* __restrict__ scores,
                             const float* __restrict__ teacher,
                             const float* __restrict__ stats,
                             float* __restrict__ out,
                             int B, int C, int G)
{
    const int tid = threadIdx.x;
    float acc = 0.0f;
    for (int row = tid; row < B; row += 256) {
        const float M = stats[row * 2 + 0];
        const float S = stats[row * 2 + 1];
        const float* t = scores + (size_t)row * (size_t)C + (size_t)row * G;
        const float* w = teacher + (size_t)row * G;
        __builtin_prefetch(t, 0, 1);
        float P = 0.0f;  // sum of exp(t_j - M) over already-masked targets
#pragma unroll 8
        for (int i = 0; i < G; ++i) {
            const float ti = t[i];
            float rem = S - P;
            rem = fmaxf(rem, 1e-37f);           // guard (16384-7 terms remain > 0)
            const float lse = M + logf(rem);
            acc += w[i] * (lse - ti);
            P += __expf(ti - M);
        }
    }
    __shared__ float red[256];
    red[tid] = acc;
    __syncthreads();
#pragma unroll
    for (int s = 128; s > 0; s >>= 1) {
        if (tid < s) red[tid] += red[tid + s];
        __syncthreads();
    }
    if (tid == 0) out[0] = red[0] / (float)B;   // sum_i mean_rows(...)
}

// ---------------------------------------------------------------------------
extern "C" void kernel_launch(void* const* d_in, const int* in_sizes, int n_in,
                              void* d_out, int out_size, void* d_ws, size_t ws_size,
                              hipStream_t stream)
{
    const float* scores  = (const float*)d_in[0];   // [B, C] f32
    const float* teacher = (const float*)d_in[1];   // [B, G] f32
    (void)n_in; (void)out_size; (void)ws_size;

    const int G = 8;                        // fixed in the reference
    const int B = in_sizes[1] / G;          // 2048
    const int C = in_sizes[0] / B;          // 16384 (= THREADS * VEC_PER_THREAD * 4)

    float* stats = (float*)d_ws;            // [B][2] floats = 16 KiB scratch
    float* out   = (float*)d_out;

    kd_row_stats_kernel<<<B, THREADS, 0, stream>>>(scores, stats, C);
    kd_loss_finalize_kernel<<<1, 256, 0, stream>>>(scores, teacher, stats, out,
                                                   B, C, G);
}